// BinaryCrossEntropyLoss_94489281195
// MI455X (gfx1250) — compile-verified
//
#include <hip/hip_runtime.h>

// Reference: out[i] = -class_weights[idx[i]] * log_sigmoid(logits[i, idx[i]//(2K), idx[i]%(2K)])
// where idx[i] comes from batch-0 targets: s = i/20, k = i%20,
//   idx = (k<10) ? pos_targets[0,s,k] : neg_targets[0,s,k-10]
// Shapes fixed by reference: B=4096, S=512, K=10, VOCAB=10000, row = S*2K = 10240.

#ifndef NUM_K
#define NUM_K 10
#define TWO_K 20
#endif

__global__ __launch_bounds__(256)
void bce_diag_gather_kernel(const float* __restrict__ logits,
                            const float* __restrict__ class_weights,
                            const int*   __restrict__ pos_targets,
                            const int*   __restrict__ neg_targets,
                            float*       __restrict__ out,
                            int B, int row_len, int vocab)
{
#if defined(__gfx1250__) && __has_builtin(__builtin_amdgcn_s_wait_tensorcnt)
    // CDNA5 split dependency counter (no-op cost: TENSORcnt is already 0).
    __builtin_amdgcn_s_wait_tensorcnt(0);
#endif

    const int i = blockIdx.x * blockDim.x + threadIdx.x;
    if (i >= B) return;

    // Warm the class_weights table (40KB) while the target load is in flight.
    // One 64B cacheline per early thread; guarded to stay inside the buffer.
    if (i * 16 + 15 < vocab) {
        __builtin_prefetch(class_weights + i * 16, 0, 3);
    }

    // idx[i] from batch-0 concatenated [pos | neg] targets.
    const int s = i / TWO_K;
    const int k = i - s * TWO_K;
    const int base = s * NUM_K;
    const int idx = (k < NUM_K) ? pos_targets[base + k]
                                : neg_targets[base + (k - NUM_K)];

    // Gather the single logit from row i (row_len = S*2K = 10240, idx < vocab <= row_len).
    const float x = logits[(long long)i * (long long)row_len + (long long)idx];

    // Numerically stable log_sigmoid, matching jax.nn.log_sigmoid = -softplus(-x):
    //   log_sigmoid(x) = min(x, 0) - log1p(exp(-|x|))
    const float ls = fminf(x, 0.0f) - log1pf(expf(-fabsf(x)));

    out[i] = -class_weights[idx] * ls;
}

extern "C" void kernel_launch(void* const* d_in, const int* in_sizes, int n_in,
                              void* d_out, int out_size, void* d_ws, size_t ws_size,
                              hipStream_t stream) {
    const float* logits        = (const float*)d_in[0]; // (B, S, 2K) f32
    const float* class_weights = (const float*)d_in[1]; // (VOCAB,)   f32
    const int*   pos_targets   = (const int*)  d_in[2]; // (B, S, K)  int
    const int*   neg_targets   = (const int*)  d_in[3]; // (B, S, K)  int
    float*       out           = (float*)d_out;         // (B,)       f32

    const int B       = out_size;              // 4096
    const int row_len = in_sizes[0] / B;       // S*2K = 10240
    const int vocab   = in_sizes[1];           // 10000

    const int threads = 256;                   // 8 wave32s per block
    const int blocks  = (B + threads - 1) / threads;
    bce_diag_gather_kernel<<<blocks, threads, 0, stream>>>(
        logits, class_weights, pos_targets, neg_targets, out, B, row_len, vocab);
}